// GradientOperator_33303176413247
// MI455X (gfx1250) — compile-verified
//
#include <hip/hip_runtime.h>
#include <stdint.h>

#define BLOCK 256

typedef float v2f __attribute__((ext_vector_type(2)));
typedef __attribute__((address_space(3))) char* lds_cptr_t;

// 32-bit LDS byte offset of a generic pointer that points into LDS.
__device__ __forceinline__ uint32_t lds_addr_of(void* p) {
    return (uint32_t)(uintptr_t)(lds_cptr_t)p;
}

// CDNA5 async data mover: copy 16B from global memory at offset goff past sbase to LDS at byte offset lds_off,
// per active lane. Tracked by ASYNCcnt.
__device__ __forceinline__ void async_copy_b128(uint32_t lds_off,
                                                const void* sbase,
                                                uint32_t goff) {
    asm volatile("global_load_async_to_lds_b128 %0, %1, %2"
                 :
                 : "v"(lds_off), "v"(goff), "s"(sbase)
                 : "memory");
}

__device__ __forceinline__ void wait_asynccnt0() {
    asm volatile("s_wait_asynccnt 0" ::: "memory");
}

__global__ __launch_bounds__(BLOCK) void GradientOperator_kernel(
    const float* __restrict__ field,      // (NV,)
    const int*   __restrict__ elements,   // (NE, 3)
    const float* __restrict__ weights,    // (NE, 3, 2)
    float*       __restrict__ out,        // (NE, 2)
    int nelem)
{
    __shared__ alignas(16) int   s_idx[BLOCK * 3];  // 3072 B = 192 x 16B
    __shared__ alignas(16) float s_w  [BLOCK * 6];  // 6144 B = 384 x 16B

    const uint32_t t = threadIdx.x;
    const uint32_t b = blockIdx.x;

    const uint32_t eBytesTotal = (uint32_t)nelem * 12u;  // <= 24 MB, fits u32
    const uint32_t wBytesTotal = (uint32_t)nelem * 24u;  // <= 48 MB, fits u32
    const uint32_t eBase = b * (BLOCK * 12u);            // byte offset of this block's indices
    const uint32_t wBase = b * (BLOCK * 24u);            // byte offset of this block's weights

    const uint32_t lds_e = lds_addr_of((void*)s_idx);
    const uint32_t lds_w = lds_addr_of((void*)s_w);

    // --- Stage indices: 192 chunks of 16B, lanes 0..191 ---
    if (t < 192u) {
        uint32_t off = eBase + t * 16u;
        uint32_t lim = eBytesTotal - 16u;
        if (off > lim) off = lim;                 // tail clamp (data still valid indices)
        async_copy_b128(lds_e + t * 16u, elements, off);
    }
    // --- Stage weights: 384 chunks of 16B, chunk t and chunk 256+t ---
    {
        uint32_t off = wBase + t * 16u;
        uint32_t lim = wBytesTotal - 16u;
        if (off > lim) off = lim;
        async_copy_b128(lds_w + t * 16u, weights, off);
    }
    if (t < 128u) {
        uint32_t c   = 256u + t;
        uint32_t off = wBase + c * 16u;
        uint32_t lim = wBytesTotal - 16u;
        if (off > lim) off = lim;
        async_copy_b128(lds_w + c * 16u, weights, off);
    }

    wait_asynccnt0();     // this wave's async copies have landed in LDS
    __syncthreads();      // all waves' copies visible to everyone

    // --- Per-element compute ---
    const int i0 = s_idx[t * 3 + 0];
    const int i1 = s_idx[t * 3 + 1];
    const int i2 = s_idx[t * 3 + 2];

    // field footprint (4 MB) stays resident in L2 -> default RT hints
    const float g0 = field[i0];
    const float g1 = field[i1];
    const float g2 = field[i2];

    const float w00 = s_w[t * 6 + 0];
    const float w01 = s_w[t * 6 + 1];
    const float w10 = s_w[t * 6 + 2];
    const float w11 = s_w[t * 6 + 3];
    const float w20 = s_w[t * 6 + 4];
    const float w21 = s_w[t * 6 + 5];

    const float o0 = fmaf(g2, w20, fmaf(g1, w10, g0 * w00));
    const float o1 = fmaf(g2, w21, fmaf(g1, w11, g0 * w01));

    const uint32_t e = b * BLOCK + t;
    if (e < (uint32_t)nelem) {
        v2f v; v.x = o0; v.y = o1;
        // write-once output stream: nontemporal 8B store
        __builtin_nontemporal_store(v, (v2f*)out + e);
    }
}

extern "C" void kernel_launch(void* const* d_in, const int* in_sizes, int n_in,
                              void* d_out, int out_size, void* d_ws, size_t ws_size,
                              hipStream_t stream) {
    const float* field    = (const float*)d_in[0];
    const int*   elements = (const int*)  d_in[1];
    const float* weights  = (const float*)d_in[2];
    float*       out      = (float*)d_out;

    const int nelem = in_sizes[1] / 3;           // elements is (NE, 3)
    const int grid  = (nelem + BLOCK - 1) / BLOCK;

    GradientOperator_kernel<<<grid, BLOCK, 0, stream>>>(field, elements, weights, out, nelem);
}